// GP_conv2D_3161095929890
// MI455X (gfx1250) — compile-verified
//
#include <hip/hip_runtime.h>
#include <hip/hip_bf16.h>
#include <math.h>

// ---- problem constants (match reference) ----
#define N_B   8
#define IH_   32
#define IW_   32
#define IC_   8
#define OC_   16
#define P_    5
#define I_DIM 72          // IC*KH*KW
#define NBLK  24          // 72 / 3 inputs per WMMA block
#define R_TOT (N_B * IH_ * IW_)   // 8192 rows
#define JIT   1e-4f

typedef __attribute__((ext_vector_type(2))) float v2f;
typedef __attribute__((ext_vector_type(8))) float v8f;

// ------------------------------------------------------------------
// Phase 1: per (o, block-of-3-inputs) build block-diagonal Kinv (16x16)
// and padded h (16), plus l^2 = softplus(raw_l)^2.
// ------------------------------------------------------------------
__global__ void gp_prep_kernel(const float* __restrict__ z,
                               const float* __restrict__ h,
                               const float* __restrict__ raw_l,
                               float* __restrict__ bk,   // [OC][NBLK][16][16]
                               float* __restrict__ hb,   // [OC][NBLK][16]
                               float* __restrict__ l2g)  // [OC][I_DIM]
{
    int t = blockIdx.x * blockDim.x + threadIdx.x;
    if (t >= OC_ * NBLK) return;
    int o = t / NBLK;
    int b = t - o * NBLK;

    float* BK = bk + (t << 8);   // 256 floats
    float* HB = hb + (t << 4);   // 16 floats

    #pragma unroll 4
    for (int k = 0; k < 256; ++k) BK[k] = 0.0f;
    HB[15] = 0.0f;

    for (int j = 0; j < 3; ++j) {
        int i = b * 3 + j;
        float x  = raw_l[o * I_DIM + i];
        float sp = logf(1.0f + __expf(x));       // softplus
        float l2 = sp * sp;
        l2g[o * I_DIM + i] = l2;

        const float* zp = z + (o * I_DIM + i) * P_;
        float zz[P_];
        #pragma unroll
        for (int p = 0; p < P_; ++p) zz[p] = zp[p];

        float A[P_][P_], Inv[P_][P_];
        #pragma unroll
        for (int p = 0; p < P_; ++p)
            #pragma unroll
            for (int q = 0; q < P_; ++q) {
                float dz = zz[p] - zz[q];
                A[p][q]  = __expf(-0.5f * dz * dz / l2) + (p == q ? JIT : 0.0f);
                Inv[p][q] = (p == q) ? 1.0f : 0.0f;
            }
        // Gauss-Jordan (SPD + jitter: stable without pivoting)
        #pragma unroll
        for (int k = 0; k < P_; ++k) {
            float piv = 1.0f / A[k][k];
            #pragma unroll
            for (int c = 0; c < P_; ++c) { A[k][c] *= piv; Inv[k][c] *= piv; }
            #pragma unroll
            for (int r = 0; r < P_; ++r) {
                if (r == k) continue;
                float f = A[r][k];
                #pragma unroll
                for (int c = 0; c < P_; ++c) { A[r][c] -= f * A[k][c]; Inv[r][c] -= f * Inv[k][c]; }
            }
        }
        const float* hp = h + (o * I_DIM + i) * P_;
        #pragma unroll
        for (int p = 0; p < P_; ++p) {
            HB[j * 5 + p] = hp[p];
            #pragma unroll
            for (int q = 0; q < P_; ++q)
                BK[(j * 5 + p) * 16 + (j * 5 + q)] = Inv[p][q];
        }
    }
}

// ------------------------------------------------------------------
// Phase 2: one wave32 per (16-row tile, o). qK = q * blockdiag(Kinv)
// via 4x v_wmma_f32_16x16x4_f32; mean = qK.h, qKq = qK.q.
// ------------------------------------------------------------------
__global__ __launch_bounds__(32)
void gp_main_kernel(const float* __restrict__ xm,
                    const float* __restrict__ xv,
                    const float* __restrict__ z,
                    const float* __restrict__ bk,
                    const float* __restrict__ hb,
                    const float* __restrict__ l2g,
                    float* __restrict__ out)
{
    __shared__ float sMu[16][I_DIM];
    __shared__ float sS2[16][I_DIM];
    __shared__ float sQ[16][16];
    __shared__ float sZ[I_DIM][P_];
    __shared__ float sL2[I_DIM];
    __shared__ float sH[NBLK][16];

    const int tile = blockIdx.x;
    const int o    = blockIdx.y;
    const int lane = threadIdx.x;
    const int m    = lane & 15;    // A-row / D-column index
    const int hi   = lane >> 4;    // half-wave select
    const int r0   = tile * 16;

    // ---- stage im2col tile (16 rows x 72 features) for mean & var ----
    for (int idx = lane; idx < 16 * I_DIM; idx += 32) {
        int row = idx / I_DIM, i = idx - row * I_DIM;
        int r   = r0 + row;
        int n   = r >> 10;             // /(32*32)
        int rem = r & 1023;
        int oh  = rem >> 5, ow = rem & 31;
        int c   = i / 9;  int k9 = i - c * 9;
        int kh  = k9 / 3, kw = k9 - kh * 3;
        int ih  = oh + kh - 1, iw = ow + kw - 1;
        float mv = 0.0f, vv = 0.0f;
        if (ih >= 0 && ih < IH_ && iw >= 0 && iw < IW_) {
            int g = ((n * IH_ + ih) * IW_ + iw) * IC_ + c;
            mv = xm[g]; vv = xv[g];
        }
        sMu[row][i] = mv;
        sS2[row][i] = vv;
    }
    for (int idx = lane; idx < I_DIM * P_; idx += 32) sZ[idx / P_][idx % P_] = z[o * I_DIM * P_ + idx];
    for (int idx = lane; idx < I_DIM; idx += 32)      sL2[idx] = l2g[o * I_DIM + idx];
    for (int idx = lane; idx < NBLK * 16; idx += 32)  sH[idx >> 4][idx & 15] = hb[o * NBLK * 16 + idx];
    __syncthreads();

    float pm[8], pv[8];
    #pragma unroll
    for (int v = 0; v < 8; ++v) { pm[v] = 0.0f; pv[v] = 0.0f; }

    for (int b = 0; b < NBLK; ++b) {
        // ---- compute this lane's 8 q values (row m, K = 4c+2*hi+{0,1}) ----
        v2f a[4];
        #pragma unroll
        for (int c = 0; c < 4; ++c) {
            float q2[2];
            #pragma unroll
            for (int s = 0; s < 2; ++s) {
                int k = 4 * c + 2 * hi + s;
                float q = 0.0f;
                if (k < 15) {
                    int il = k / 5, p = k - il * 5;
                    int i  = b * 3 + il;
                    float l2v  = sL2[i];
                    float d    = l2v + sS2[m][i];
                    float diff = sMu[m][i] - sZ[i][p];
                    q = sqrtf(l2v / d) * __expf(-0.5f * diff * diff / d);
                }
                q2[s] = q;
                sQ[m][k] = q;          // tile copy for D-layout read
            }
            a[c].x = q2[0];
            a[c].y = q2[1];
        }
        __syncthreads();

        // ---- qK tile: 4 chained f32 WMMAs, B = block-diag Kinv ----
        const float* bkp = bk + (((o * NBLK) + b) << 8);
        v8f acc = {0.f, 0.f, 0.f, 0.f, 0.f, 0.f, 0.f, 0.f};
        #pragma unroll
        for (int c = 0; c < 4; ++c) {
            v2f bv;
            bv.x = bkp[(4 * c + 2 * hi + 0) * 16 + m];
            bv.y = bkp[(4 * c + 2 * hi + 1) * 16 + m];
            acc = __builtin_amdgcn_wmma_f32_16x16x4_f32(
                false, a[c], false, bv, (short)0, acc, false, false);
        }

        // ---- per-lane partial dots in D layout (M = v+8*hi, N = m) ----
        #pragma unroll
        for (int v = 0; v < 8; ++v) {
            float d  = acc[v];
            float qd = sQ[v + 8 * hi][m];
            pm[v] += d * sH[b][m];
            pv[v] += d * qd;
        }
        __syncthreads();   // before sQ is overwritten next block
    }

    // ---- reduce over N (16 lanes within each half-wave) ----
    #pragma unroll
    for (int v = 0; v < 8; ++v) {
        float a_ = pm[v], b_ = pv[v];
        #pragma unroll
        for (int off = 8; off >= 1; off >>= 1) {
            a_ += __shfl_xor(a_, off, 32);
            b_ += __shfl_xor(b_, off, 32);
        }
        pm[v] = a_; pv[v] = b_;
    }

    if (m == 0) {
        #pragma unroll
        for (int v = 0; v < 8; ++v) {
            int r = r0 + v + 8 * hi;
            out[r * OC_ + o] = pm[v];
            float var = (float)I_DIM - pv[v];
            out[R_TOT * OC_ + r * OC_ + o] = var > 1e-6f ? var : 1e-6f;
        }
    }
}

// ------------------------------------------------------------------
extern "C" void kernel_launch(void* const* d_in, const int* in_sizes, int n_in,
                              void* d_out, int out_size, void* d_ws, size_t ws_size,
                              hipStream_t stream) {
    const float* xm = (const float*)d_in[0];  // x_mean [8,32,32,8]
    const float* xv = (const float*)d_in[1];  // x_var  [8,32,32,8]
    const float* z  = (const float*)d_in[2];  // [16,72,5]
    const float* h  = (const float*)d_in[3];  // [16,72,5]
    const float* rl = (const float*)d_in[4];  // [16,72]
    float* out = (float*)d_out;               // mean flat then var flat

    float* ws  = (float*)d_ws;
    float* bk  = ws;                               // 16*24*256 = 98304 floats
    float* hb  = ws + OC_ * NBLK * 256;            // 16*24*16  =  6144 floats
    float* l2g = hb + OC_ * NBLK * 16;             // 16*72     =  1152 floats

    gp_prep_kernel<<<6, 64, 0, stream>>>(z, h, rl, bk, hb, l2g);

    dim3 grid(R_TOT / 16, OC_);   // 512 x 16
    gp_main_kernel<<<grid, 32, 0, stream>>>(xm, xv, z, bk, hb, l2g, out);
}